// DisMult_13013750907165
// MI455X (gfx1250) — compile-verified
//
#include <hip/hip_runtime.h>
#include <hip/hip_bf16.h>

typedef __attribute__((ext_vector_type(2))) float v2f;
typedef __attribute__((ext_vector_type(8))) float v8f;

#define DIM 128
#define N_REL 3
#define N_EDGES 500000
#define EDGES_PER_TILE 16
#define TILES_PER_REL (N_EDGES / EDGES_PER_TILE)   // 31250
#define TOTAL_TILES (N_REL * TILES_PER_REL)        // 93750
#define WAVES_PER_BLOCK 8

// One wave32 handles 16 edges of one relation.
// A = xs (*) rel  (16 x 4 fp32 slice), B = xd (4 x 16 fp32 slice),
// C accumulates 16x16; per-edge DistMult scores are diag(C).
__global__ __launch_bounds__(WAVES_PER_BLOCK * 32)
void distmult_wmma_f32(const float* __restrict__ node_embeds,
                       const float* __restrict__ rel_emb,
                       const int*   __restrict__ src_idx,
                       const int*   __restrict__ dst_idx,
                       float*       __restrict__ out)
{
    const int lane = threadIdx.x & 31;
    const int wave = threadIdx.x >> 5;
    const int tile = blockIdx.x * WAVES_PER_BLOCK + wave;
    if (tile >= TOTAL_TILES) return;   // whole-wave exit; EXEC stays all-1s otherwise

    const int r = tile / TILES_PER_REL;
    const int t = tile % TILES_PER_REL;
    const long ebase = (long)t * EDGES_PER_TILE;

    // A-matrix layout (32-bit, 16x4): lanes 0-15 carry M=lane, K = k0+{0,1};
    // lanes 16-31 carry M=lane-16, K = k0+{2,3}. B mirrors with N = lane&15.
    const int m    = lane & 15;          // edge-in-tile this lane serves
    const int koff = (lane >> 4) * 2;    // 0 for lanes 0-15, 2 for lanes 16-31

    const long eidx = (long)r * N_EDGES + ebase + m;
    const long srow = (long)src_idx[eidx] * DIM;
    const long drow = (long)dst_idx[eidx] * DIM;

    const float* __restrict__ sp = node_embeds + srow + koff;
    const float* __restrict__ dp = node_embeds + drow + koff;
    const float* __restrict__ rp = rel_emb + r * DIM + koff;

    v8f c = {};
#pragma unroll
    for (int k0 = 0; k0 < DIM; k0 += 4) {
        v2f s  = *(const v2f*)(sp + k0);
        v2f d  = *(const v2f*)(dp + k0);
        v2f rv = *(const v2f*)(rp + k0);
        v2f a;
        a.x = s.x * rv.x;
        a.y = s.y * rv.y;
        // D = A(16x4) * B(4x16) + C   -> v_wmma_f32_16x16x4_f32
        c = __builtin_amdgcn_wmma_f32_16x16x4_f32(
                /*neg_a=*/false, a, /*neg_b=*/false, d,
                /*c_mod=*/(short)0, c, /*reuse_a=*/false, /*reuse_b=*/false);
    }

    // Diagonal extraction:
    //   m in 0..7  : C[m][m]  lives in vgpr m,   lane m
    //   m in 8..15 : C[m][m]  lives in vgpr m-8, lane m+16 (lanes 24..31)
    float sc = 0.0f;
#pragma unroll
    for (int j = 0; j < 8; ++j) {
        bool sel = (lane == j) || (lane == j + 24);
        sc = sel ? c[j] : sc;
    }

    float* orow = out + (long)r * N_EDGES + ebase;
    if (lane < 8) {
        orow[lane] = sc;                 // edges 0..7
    } else if (lane >= 24) {
        orow[lane - 16] = sc;            // edges 8..15
    }
}

extern "C" void kernel_launch(void* const* d_in, const int* in_sizes, int n_in,
                              void* d_out, int out_size, void* d_ws, size_t ws_size,
                              hipStream_t stream)
{
    const float* node_embeds = (const float*)d_in[0];  // [100000, 128] fp32
    const float* rel_emb     = (const float*)d_in[1];  // [3, 128] fp32
    const int*   src_idx     = (const int*)d_in[2];    // [3, 500000] int
    const int*   dst_idx     = (const int*)d_in[3];    // [3, 500000] int
    float*       out         = (float*)d_out;          // [3, 500000] fp32

    const int blocks = (TOTAL_TILES + WAVES_PER_BLOCK - 1) / WAVES_PER_BLOCK;
    distmult_wmma_f32<<<blocks, WAVES_PER_BLOCK * 32, 0, stream>>>(
        node_embeds, rel_emb, src_idx, dst_idx, out);
}